// FocalLoss_37589553774631
// MI455X (gfx1250) — compile-verified
//
#include <hip/hip_runtime.h>
#include <hip/hip_bf16.h>

#define ALPHA_F   0.25f
#define NEGA_F    0.75f      // 1 - ALPHA
#define LAMDA_F   50.0f
#define POS_T_F   0.5f
#define NEG_T_F   0.4f
#define BETA_F    (1.0f/9.0f)
#define CLIP_F    1.0e-4f

#define B_IMG  8
#define N_ANC  100000
#define M_BOX  32
#define C_CLS  80
#define NBLK   1024                 // focal partial blocks per image (multiple of 64)
#define EPI    (N_ANC * C_CLS / 4)  // float4 elements per image = 2,000,000

typedef float v2f __attribute__((ext_vector_type(2)));
typedef float v4f __attribute__((ext_vector_type(4)));
typedef float v8f __attribute__((ext_vector_type(8)));

__device__ __forceinline__ float waveReduce(float v) {
    // wave32 butterfly reduction
    for (int m = 16; m >= 1; m >>= 1) v += __shfl_xor(v, m, 32);
    return v;
}

// ---------------------------------------------------------------- k_zero
__global__ void k_zero(float* regs, float* npos) {
    int t = threadIdx.x;
    if (t < B_IMG) { regs[t] = 0.0f; npos[t] = 0.0f; }
}

// ---------------------------------------------------------------- k_assign
// One thread per (image, anchor). Boxes/labels staged in LDS.
__global__ void k_assign(const float4* __restrict__ anc,
                         const float4* __restrict__ regp,
                         const float*  __restrict__ boxes,
                         const int*    __restrict__ labels,
                         int* __restrict__ meta,
                         float* __restrict__ regs,
                         float* __restrict__ npos) {
    __shared__ float sB[M_BOX * 4];
    __shared__ int   sL[M_BOX];
    __shared__ float sredR[8];
    __shared__ float sredP[8];

    const int b   = blockIdx.y;
    const int tid = threadIdx.x;
    if (tid < M_BOX * 4) sB[tid] = boxes[b * M_BOX * 4 + tid];
    if (tid < M_BOX)     sL[tid] = labels[b * M_BOX + tid];
    __syncthreads();

    const int n = blockIdx.x * blockDim.x + tid;
    float myreg = 0.0f, mypos = 0.0f;

    if (n < N_ANC) {
        float4 a = anc[n];
        float areaA = (a.z - a.x) * (a.w - a.y);
        float best = -1.0f; int arg = 0;
        #pragma unroll 4
        for (int i = 0; i < M_BOX; ++i) {
            float bx1 = sB[i*4+0], by1 = sB[i*4+1], bx2 = sB[i*4+2], by2 = sB[i*4+3];
            float iw = fminf(a.z, bx2) - fmaxf(a.x, bx1);
            float ih = fminf(a.w, by2) - fmaxf(a.y, by1);
            iw = fmaxf(iw, 0.0f); ih = fmaxf(ih, 0.0f);
            float inter = iw * ih;
            float areaB = (bx2 - bx1) * (by2 - by1);
            float uni   = fmaxf(areaA + areaB - inter, 1e-8f);
            float iou   = inter / uni;
            if (iou > best) { best = iou; arg = i; }   // first-occurrence argmax
        }
        bool pos = best >= POS_T_F;
        bool neg = best <  NEG_T_F;
        int  m   = pos ? sL[arg] : (neg ? -1 : -2);
        meta[b * N_ANC + n] = m;

        if (pos) {
            float bx1 = sB[arg*4+0], by1 = sB[arg*4+1], bx2 = sB[arg*4+2], by2 = sB[arg*4+3];
            float bw = fmaxf(bx2 - bx1, 1.0f), bh = fmaxf(by2 - by1, 1.0f);
            float bcx = bx1 + bw * 0.5f, bcy = by1 + bh * 0.5f;
            float aw = fmaxf(a.z - a.x, 1.0f), ah = fmaxf(a.w - a.y, 1.0f);
            float acx = a.x + aw * 0.5f, acy = a.y + ah * 0.5f;
            // (d - MEAN)/STD ; MEAN = 0, STD = [.1,.1,.2,.2]
            float t0 = ((bcx - acx) / aw) * 10.0f;
            float t1 = ((bcy - acy) / ah) * 10.0f;
            float t2 = __logf(bw / aw) * 5.0f;
            float t3 = __logf(bh / ah) * 5.0f;
            float4 r = regp[b * N_ANC + n];
            float s = 0.0f, d;
            d = fabsf(r.x - t0); s += (d <= BETA_F) ? 4.5f*d*d : d - 0.5f*BETA_F;
            d = fabsf(r.y - t1); s += (d <= BETA_F) ? 4.5f*d*d : d - 0.5f*BETA_F;
            d = fabsf(r.z - t2); s += (d <= BETA_F) ? 4.5f*d*d : d - 0.5f*BETA_F;
            d = fabsf(r.w - t3); s += (d <= BETA_F) ? 4.5f*d*d : d - 0.5f*BETA_F;
            myreg = s; mypos = 1.0f;
        }
    }

    // block reduce (8 wave32 waves)
    const int wave = tid >> 5, lane = tid & 31;
    float wr = waveReduce(myreg);
    float wp = waveReduce(mypos);
    if (lane == 0) { sredR[wave] = wr; sredP[wave] = wp; }
    __syncthreads();
    if (tid == 0) {
        float tr = 0.0f, tp = 0.0f;
        #pragma unroll
        for (int i = 0; i < 8; ++i) { tr += sredR[i]; tp += sredP[i]; }
        if (tp != 0.0f) { atomicAdd(&regs[b], tr); atomicAdd(&npos[b], tp); }
    }
}

// ---------------------------------------------------------------- k_focal
// Streaming pass over cls_preds (256 MB): coalesced float4 NT loads (single-use
// stream larger than the 192 MB L2 -> don't pollute cache), one log/element.
__global__ void k_focal(const v4f* __restrict__ cls,
                        const int* __restrict__ meta,
                        float* __restrict__ partials) {
    __shared__ float sred[8];
    const int b   = blockIdx.y;
    const int tid = threadIdx.x;
    const v4f* __restrict__ src = cls + (size_t)b * EPI;
    const int* __restrict__ mb  = meta + b * N_ANC;

    const int stride = NBLK * 256;
    float acc = 0.0f;
    for (int idx = blockIdx.x * 256 + tid; idx < EPI; idx += stride) {
        __builtin_prefetch(&src[idx + stride], 0, 0);          // global_prefetch_b8
        v4f v = __builtin_nontemporal_load(&src[idx]);         // b128 th:NT
        int anchor = idx / 20;          // 20 float4 per anchor (C=80)
        int c0     = (idx % 20) * 4;
        int m      = mb[anchor];
        if (m != -2) {
            float p, fl = 0.0f;
            p = fminf(fmaxf(v[0], CLIP_F), 1.0f - CLIP_F); fl += -NEGA_F * p*p * __logf(1.0f - p);
            p = fminf(fmaxf(v[1], CLIP_F), 1.0f - CLIP_F); fl += -NEGA_F * p*p * __logf(1.0f - p);
            p = fminf(fmaxf(v[2], CLIP_F), 1.0f - CLIP_F); fl += -NEGA_F * p*p * __logf(1.0f - p);
            p = fminf(fmaxf(v[3], CLIP_F), 1.0f - CLIP_F); fl += -NEGA_F * p*p * __logf(1.0f - p);
            unsigned j = (unsigned)(m - c0);
            if (j < 4u) {  // rare: replace the matched-class neg term with the pos term
                float pv = v[j];
                p = fminf(fmaxf(pv, CLIP_F), 1.0f - CLIP_F);
                float negt = -NEGA_F  * p * p * __logf(1.0f - p);
                float post = -ALPHA_F * (1.0f - p) * (1.0f - p) * __logf(p);
                fl += post - negt;
            }
            acc += fl;
        }
    }

    const int wave = tid >> 5, lane = tid & 31;
    float w = waveReduce(acc);
    if (lane == 0) sred[wave] = w;
    __syncthreads();
    if (tid == 0) {
        float t = 0.0f;
        #pragma unroll
        for (int i = 0; i < 8; ++i) t += sred[i];
        partials[b * NBLK + blockIdx.x] = t;
    }
}

// ---------------------------------------------------------------- k_final
// One wave32. WMMA f32 16x16x4 with A=ones accumulates 64 partials per
// instruction: sum over all 256 C slots == 16 * sum(inputs)  (layout-agnostic).
__global__ void k_final(const float* __restrict__ partials,
                        const float* __restrict__ regs,
                        const float* __restrict__ npos,
                        float* __restrict__ out) {
    const int lane = threadIdx.x;   // 32 threads = one full wave (EXEC all ones)
    float cls_acc = 0.0f, reg_acc = 0.0f;

    for (int b = 0; b < B_IMG; ++b) {
        const float* __restrict__ P = partials + b * NBLK;
        float total;
#if defined(__gfx1250__) && __has_builtin(__builtin_amdgcn_wmma_f32_16x16x4_f32)
        v8f c = {0.f,0.f,0.f,0.f,0.f,0.f,0.f,0.f};
        v2f ones; ones.x = 1.0f; ones.y = 1.0f;
        for (int i = 0; i < NBLK; i += 64) {
            v2f bv; bv.x = P[i + lane]; bv.y = P[i + 32 + lane];
            // (neg_a, A, neg_b, B, c_mod, C, reuse_a, reuse_b)
            c = __builtin_amdgcn_wmma_f32_16x16x4_f32(false, ones, false, bv,
                                                      (short)0, c, false, false);
        }
        float s = c[0]+c[1]+c[2]+c[3]+c[4]+c[5]+c[6]+c[7];
        s = waveReduce(s);
        total = s * (1.0f / 16.0f);
#else
        float s = 0.0f;
        for (int i = lane; i < NBLK; i += 32) s += P[i];
        total = waveReduce(s);
#endif
        if (lane == 0) {
            float np = npos[b];
            cls_acc += total / fmaxf(np, 1.0f);
            reg_acc += (np > 0.0f) ? regs[b] / fmaxf(np * 4.0f, 1.0f) : 0.0f;
        }
    }
    if (lane == 0) {
        out[0] = cls_acc / (float)B_IMG;
        out[1] = LAMDA_F * reg_acc / (float)B_IMG;
    }
}

// ---------------------------------------------------------------- launch
extern "C" void kernel_launch(void* const* d_in, const int* in_sizes, int n_in,
                              void* d_out, int out_size, void* d_ws, size_t ws_size,
                              hipStream_t stream) {
    const v4f*    cls    = (const v4f*)   d_in[0];   // [B,N,C] f32
    const float4* regp   = (const float4*)d_in[1];   // [B,N,4] f32
    const float4* anc    = (const float4*)d_in[2];   // [1,N,4] f32
    const float*  boxes  = (const float*) d_in[3];   // [B,M,4] f32
    const int*    labels = (const int*)   d_in[4];   // [B,M]   i32
    float* out = (float*)d_out;

    // workspace layout
    float* partials = (float*)d_ws;                  // B*NBLK floats (32 KB)
    float* regs     = partials + B_IMG * NBLK;       // B floats
    float* npos     = regs + B_IMG;                  // B floats
    int*   meta     = (int*)(npos + B_IMG);          // B*N ints (3.2 MB)

    k_zero  <<<1, 32, 0, stream>>>(regs, npos);
    k_assign<<<dim3((N_ANC + 255) / 256, B_IMG), 256, 0, stream>>>(
        anc, regp, boxes, labels, meta, regs, npos);
    k_focal <<<dim3(NBLK, B_IMG), 256, 0, stream>>>(cls, meta, partials);
    k_final <<<1, 32, 0, stream>>>(partials, regs, npos, out);
}